// Model_87668872446143
// MI455X (gfx1250) — compile-verified
//
#include <hip/hip_runtime.h>
#include <hip/hip_bf16.h>

typedef __attribute__((ext_vector_type(2))) float v2f;
typedef __attribute__((ext_vector_type(8))) float v8f;

#define RES   100
#define DIN   24
#define SEG   48
#define NPAD  112            // 100 rounded up to 7*16
#define S_TOT 161280L        // (512*720*21)/48

// ---------------------------------------------------------------------------
// selu: scale * (x > 0 ? x : alpha*(exp(x)-1))
// ---------------------------------------------------------------------------
__device__ __forceinline__ float selu_f(float x) {
    const float scale = 1.0507009873554805f;
    const float alpha = 1.6732632423543772f;
    float neg = scale * alpha * (__expf(x) - 1.0f);
    return x > 0.0f ? scale * x : neg;
}

// ---------------------------------------------------------------------------
// Kernel 1: output permute  x(512,720,21) -> out(512,21,720)
// Tile of 24 time-steps: x[b][t0..t0+23][0..20] is 504 contiguous floats.
// Coalesced load into LDS, coalesced (t-fastest) store.
// ---------------------------------------------------------------------------
__global__ void k_transpose(const float* __restrict__ x, float* __restrict__ out) {
    __shared__ float tile[24 * 21];
    int b  = blockIdx.x / 30;            // 720 / 24 = 30 tiles per batch
    int t0 = (blockIdx.x % 30) * 24;
    int tid = threadIdx.x;               // 0..503
    tile[tid] = x[(long)b * (720 * 21) + (long)t0 * 21 + tid];
    __syncthreads();
    int c = tid / 24;                    // channel 0..20
    int i = tid % 24;                    // t offset (fastest -> coalesced)
    out[(long)b * (21 * 720) + (long)c * 720 + (t0 + i)] = tile[i * 21 + c];
}

// ---------------------------------------------------------------------------
// Kernel 2: W_comb[NPAD][48] = W_in(100x24) @ W_proj(24x48), zero-padded rows
// ---------------------------------------------------------------------------
__global__ void k_wcomb(const float* __restrict__ Wp,   // (24,48)
                        const float* __restrict__ Wi,   // (100,24)
                        float* __restrict__ Wc) {       // (NPAD,48)
    int idx = blockIdx.x * blockDim.x + threadIdx.x;
    if (idx >= NPAD * SEG) return;
    int n = idx / SEG, k = idx % SEG;
    float acc = 0.0f;
    if (n < RES) {
        #pragma unroll
        for (int j = 0; j < DIN; ++j)
            acc += Wi[n * DIN + j] * Wp[j * SEG + k];
    }
    Wc[idx] = acc;
}

// ---------------------------------------------------------------------------
// Kernel 3: U[r][n] = X[t0+r][0..47] . Wc[n][0..47]   via V_WMMA_F32_16X16X4_F32
// One wave -> one 16-row tile x all 7 N-tiles. 4 waves / block (64 rows).
// A layout (16x4 f32): lane&15 = M, lane>>4 picks K {0,1} vs {2,3} in v2f.
// B layout mirrors A with N in place of M. C/D: v8f, lanes 0-15 = rows 0-7.
// ---------------------------------------------------------------------------
__global__ void k_gemm_wmma(const float* __restrict__ X,   // flat x, rows of 48
                            const float* __restrict__ Wc,  // (NPAD,48)
                            float* __restrict__ U,         // (rows, NPAD) chunk
                            long t0) {
    int wave = threadIdx.x >> 5;
    int lane = threadIdx.x & 31;
    int m    = lane & 15;
    int g    = lane >> 4;                          // 0: K={0,1}, 1: K={2,3}
    long tile_row = (long)(blockIdx.x * 4 + wave) * 16;

    // A: 16 rows x 48 cols of X, 12 k-blocks of v2f per lane
    const float* arow = X + (t0 + tile_row + m) * SEG + g * 2;
    v2f a[12];
    #pragma unroll
    for (int kb = 0; kb < 12; ++kb)
        a[kb] = *(const v2f*)(arow + kb * 4);

    #pragma unroll
    for (int nt = 0; nt < 7; ++nt) {
        const float* brow = Wc + (nt * 16 + m) * SEG + g * 2;
        v8f acc = {};
        #pragma unroll
        for (int kb = 0; kb < 12; ++kb) {
            v2f b = *(const v2f*)(brow + kb * 4);
            acc = __builtin_amdgcn_wmma_f32_16x16x4_f32(
                false, a[kb], false, b, (short)0, acc, false, false);
        }
        // store 16x16 tile: lane holds column (nt*16+m), rows g*8 .. g*8+7
        float* uo = U + (tile_row + g * 8) * NPAD + nt * 16 + m;
        #pragma unroll
        for (int v = 0; v < 8; ++v)
            uo[(long)v * NPAD] = acc[v];
    }
}

// ---------------------------------------------------------------------------
// Kernel 4: sequential reservoir update, one workgroup, lane = state component.
// s_t = selu(u_t + d*s_{t-1}); 8-deep register prefetch ring hides L2 latency.
// R is a multiple of 64 (so of 8, and >= 64).
// ---------------------------------------------------------------------------
__global__ void k_reservoir(const float* __restrict__ U,    // (R, NPAD)
                            const float* __restrict__ d,
                            float* __restrict__ state,      // NPAD persisted
                            float* __restrict__ out_tail,   // 100 floats
                            int R, int first, int last) {
    int tid = threadIdx.x;
    if (tid >= NPAD) return;
    float dv = (tid < RES) ? d[tid] : 0.0f;
    float s  = first ? 0.0f : state[tid];

    const float* u = U + tid;
    float b0 = u[0 * NPAD], b1 = u[1 * NPAD], b2 = u[2 * NPAD], b3 = u[3 * NPAD];
    float b4 = u[4 * NPAD], b5 = u[5 * NPAD], b6 = u[6 * NPAD], b7 = u[7 * NPAD];

    const float* pref = u + 8 * NPAD;
    int t = 0;
    for (; t + 16 <= R; t += 8) {
        s = selu_f(b0 + dv * s); b0 = pref[0 * NPAD];
        s = selu_f(b1 + dv * s); b1 = pref[1 * NPAD];
        s = selu_f(b2 + dv * s); b2 = pref[2 * NPAD];
        s = selu_f(b3 + dv * s); b3 = pref[3 * NPAD];
        s = selu_f(b4 + dv * s); b4 = pref[4 * NPAD];
        s = selu_f(b5 + dv * s); b5 = pref[5 * NPAD];
        s = selu_f(b6 + dv * s); b6 = pref[6 * NPAD];
        s = selu_f(b7 + dv * s); b7 = pref[7 * NPAD];
        pref += 8 * NPAD;
    }
    // epilogue: last 8 steps, no refill
    s = selu_f(b0 + dv * s);
    s = selu_f(b1 + dv * s);
    s = selu_f(b2 + dv * s);
    s = selu_f(b3 + dv * s);
    s = selu_f(b4 + dv * s);
    s = selu_f(b5 + dv * s);
    s = selu_f(b6 + dv * s);
    s = selu_f(b7 + dv * s);

    state[tid] = s;
    if (last && tid < RES) out_tail[tid] = s;
}

// ---------------------------------------------------------------------------
// Launcher
// ---------------------------------------------------------------------------
extern "C" void kernel_launch(void* const* d_in, const int* in_sizes, int n_in,
                              void* d_out, int out_size, void* d_ws, size_t ws_size,
                              hipStream_t stream) {
    const float* x  = (const float*)d_in[0];   // 512*720*21
    const float* Wp = (const float*)d_in[1];   // 24*48
    const float* Wi = (const float*)d_in[2];   // 100*24
    const float* dd = (const float*)d_in[3];   // 100
    float* out      = (float*)d_out;           // 7741440 + 100

    float* ws    = (float*)d_ws;
    float* Wc    = ws;                 // NPAD*48 = 5376 floats
    float* state = ws + 5376;          // NPAD floats (persists between chunks)
    float* Ubuf  = ws + 5504;          // chunked U (rows x NPAD)

    // 1) output permute (independent of everything else)
    k_transpose<<<512 * 30, 504, 0, stream>>>(x, out);

    // 2) fold the two projection matrices
    k_wcomb<<<(NPAD * SEG + 255) / 256, 256, 0, stream>>>(Wp, Wi, Wc);

    // 3+4) chunked GEMM -> recurrence pipeline (stream order = dependency)
    long cap = ((long)(ws_size / sizeof(float)) - 5504) / NPAD;
    long chunk = cap - (cap % 64);
    if (chunk > S_TOT) chunk = S_TOT;
    if (chunk < 64)    chunk = 64;     // assumes ws holds at least one tile row-block

    float* out_tail = out + (out_size - RES);
    long t0 = 0;
    while (t0 < S_TOT) {
        long r = S_TOT - t0;
        if (r > chunk) r = chunk;      // S_TOT % 64 == 0 and chunk % 64 == 0 -> r % 64 == 0
        k_gemm_wmma<<<(int)(r / 64), 128, 0, stream>>>(x, Wc, Ubuf, t0);
        k_reservoir<<<1, 128, 0, stream>>>(Ubuf, dd, state, out_tail,
                                           (int)r, t0 == 0, (t0 + r) == S_TOT);
        t0 += r;
    }
}